// SocialLSTM_77970836291685
// MI455X (gfx1250) — compile-verified
//
#include <hip/hip_runtime.h>
#include <math.h>

// Problem constants (from reference)
#define TT 20
#define NN 1024
#define G2 16
#define RR 32
#define EE 64
#define OO 5

#define SPLIT 4            // k-dimension (m) split for social kernel
#define MCHUNK (NN / SPLIT)  // 256 rows per chunk

typedef float v2f __attribute__((ext_vector_type(2)));
typedef float v8f __attribute__((ext_vector_type(8)));

// ---------------------------------------------------------------------------
// Kernel 1: social tensor partials via V_WMMA_F32_16X16X4_F32.
// social[n, g, r] = sum_m grid[n, m, g] * h[m, r]   (per node: [16x1024]x[1024x32])
// Grid: (N/16, SPLIT) blocks of 512 threads (16 waves). Wave w computes node
// blockIdx.x*16+w over m-range [chunk*256, chunk*256+256), accumulating into
// two 16x16 f32 tiles, stored to partials[chunk][node][g*32+r].
// A lane map (32-bit A 16x4): lanes 0-15 M=lane K={0,1}; lanes 16-31 M=lane-16
// K={2,3}. B (4x16) symmetric. D: lanes 0-15 N=lane M=v; lanes 16-31 M=8+v.
// grid_t fp32 streamed once from HBM -> bandwidth-bound; h slice in LDS (32KB).
// ---------------------------------------------------------------------------
__global__ void social_wmma_kernel(const float* __restrict__ grid_t,   // [N, N, G2]
                                   const float* __restrict__ h,        // [N, R]
                                   float* __restrict__ partials)       // [SPLIT, N, G2*R]
{
    extern __shared__ float s_h[];   // MCHUNK * RR floats = 32 KB

    const int tid   = threadIdx.x;   // 512 threads = 16 waves
    const int chunk = blockIdx.y;
    const int m0    = chunk * MCHUNK;

    // stage this chunk's h rows into LDS (float4 coalesced)
    {
        const float4* h4 = (const float4*)(h + (size_t)m0 * RR);
        float4* s4 = (float4*)s_h;
        #pragma unroll
        for (int i = tid; i < (MCHUNK * RR) / 4; i += 512) s4[i] = h4[i];
    }
    __syncthreads();

    const int wave = tid >> 5;
    const int lane = tid & 31;
    const int node = blockIdx.x * 16 + wave;

    const int g  = lane & 15;          // A row (g) / B-D col (r)
    const int kb = (lane >> 4) << 1;   // K base: 0 (lanes 0-15) or 2 (16-31)

    const float* gA = grid_t + ((size_t)node * NN + m0) * G2;  // rows of 16 floats

    v8f acc0 = {};   // r = 0..15
    v8f acc1 = {};   // r = 16..31

    #pragma unroll 4
    for (int m = 0; m < MCHUNK; m += 4) {
        const float* arow = gA + (size_t)(m + kb) * G2 + g;
        // prefetch grid stream ~1KB (16 rows) ahead; speculative, safe at tail
        __builtin_prefetch(arow + 16 * G2, 0, 1);

        v2f a;
        a.x = arow[0];        // grid[n, m0+m+kb,   g]
        a.y = arow[G2];       // grid[n, m0+m+kb+1, g]

        const float* brow = s_h + (m + kb) * RR;
        v2f b0, b1;
        b0.x = brow[g];        b0.y = brow[RR + g];        // h[.., r]
        b1.x = brow[16 + g];   b1.y = brow[RR + 16 + g];   // h[.., r+16]

        acc0 = __builtin_amdgcn_wmma_f32_16x16x4_f32(false, a, false, b0,
                                                     (short)0, acc0, false, false);
        acc1 = __builtin_amdgcn_wmma_f32_16x16x4_f32(false, a, false, b1,
                                                     (short)0, acc1, false, false);
    }

    // store D tile: partials[chunk, n, g_out*32 + r], g_out = 8*(lane>>4)+v
    float* outp = partials + ((size_t)chunk * NN + node) * (G2 * RR);
    const int mh = (lane >> 4) * 8;
    #pragma unroll
    for (int v = 0; v < 8; ++v) {
        outp[(mh + v) * RR + g]      = acc0[v];
        outp[(mh + v) * RR + 16 + g] = acc1[v];
    }
}

// ---------------------------------------------------------------------------
// Kernel 1b: reduce SPLIT partials -> social.  float4, pure bandwidth (~10MB).
// ---------------------------------------------------------------------------
__global__ void social_reduce_kernel(const float* __restrict__ partials,
                                     float* __restrict__ social)
{
    const int i = blockIdx.x * blockDim.x + threadIdx.x;   // float4 index
    const size_t S = (size_t)NN * G2 * RR / 4;             // per-chunk float4 count
    const float4* p = (const float4*)partials;
    float4 a = p[i];
    float4 b = p[i + S];
    float4 c = p[i + 2 * S];
    float4 d = p[i + 3 * S];
    float4 r;
    r.x = (a.x + b.x) + (c.x + d.x);
    r.y = (a.y + b.y) + (c.y + d.y);
    r.z = (a.z + b.z) + (c.z + d.z);
    r.w = (a.w + b.w) + (c.w + d.w);
    ((float4*)social)[i] = r;
}

// ---------------------------------------------------------------------------
// Kernel 2: fused embeddings + gates + LSTM pointwise + output projection.
// One block (256 threads) handles 64 nodes. W_soc staged in LDS.
// ~55 MMAC/step total -> negligible vs social kernel; VALU is fine here.
// ---------------------------------------------------------------------------
#define NB 64  // nodes per block

__device__ __forceinline__ float sigmoidf_(float x) { return 1.0f / (1.0f + __expf(-x)); }

__global__ void lstm_step_kernel(const float* __restrict__ x_t,     // [N, 2]
                                 const float* __restrict__ social,  // [N, 512]
                                 float* __restrict__ h,             // [N, R] in/out
                                 float* __restrict__ c,             // [N, R] in/out
                                 const float* __restrict__ W_in,    // [2, E]
                                 const float* __restrict__ b_in,    // [E]
                                 const float* __restrict__ W_soc,   // [512, E]
                                 const float* __restrict__ b_soc,   // [E]
                                 const float* __restrict__ W_ih,    // [128, 128]
                                 const float* __restrict__ W_hh,    // [32, 128]
                                 const float* __restrict__ b_lstm,  // [128]
                                 const float* __restrict__ W_out,   // [32, 5]
                                 const float* __restrict__ b_out,   // [5]
                                 float* __restrict__ out_t)         // [N, O]
{
    extern __shared__ float smem[];
    float* s_Wsoc  = smem;                    // 512*64 = 32768
    float* s_iemb  = s_Wsoc + 512 * EE;       // 64*64  =  4096
    float* s_temb  = s_iemb + NB * EE;        // 64*64  =  4096
    float* s_gates = s_temb + NB * EE;        // 64*128 =  8192
    float* s_hnew  = s_gates + NB * 128;      // 64*32  =  2048

    const int tid   = threadIdx.x;
    const int node0 = blockIdx.x * NB;

    // stage W_soc (128KB) into LDS, float4 coalesced
    {
        const float4* w4 = (const float4*)W_soc;
        float4* s4 = (float4*)s_Wsoc;
        #pragma unroll 4
        for (int i = tid; i < (512 * EE) / 4; i += 256) s4[i] = w4[i];
    }
    __syncthreads();

    // Phase 1: input_emb and tensor_emb  (64 nodes x 64 e)
    for (int w = tid; w < NB * EE; w += 256) {
        const int nl = w >> 6;
        const int e  = w & 63;
        const int n  = node0 + nl;

        float ie = b_in[e] + x_t[n * 2 + 0] * W_in[e] + x_t[n * 2 + 1] * W_in[EE + e];
        s_iemb[nl * EE + e] = ie > 0.0f ? ie : 0.0f;

        const float* soc = social + (size_t)n * (G2 * RR);
        float te = b_soc[e];
        #pragma unroll 8
        for (int k = 0; k < G2 * RR; ++k) te += soc[k] * s_Wsoc[k * EE + e];
        s_temb[nl * EE + e] = te > 0.0f ? te : 0.0f;
    }
    __syncthreads();

    // Phase 2: gates = [iemb,temb] @ W_ih + h @ W_hh + b  (64 nodes x 128 j)
    for (int w = tid; w < NB * 128; w += 256) {
        const int nl = w >> 7;
        const int j  = w & 127;
        const int n  = node0 + nl;

        float acc = b_lstm[j];
        #pragma unroll 8
        for (int k = 0; k < EE; ++k) acc += s_iemb[nl * EE + k] * W_ih[k * 128 + j];
        #pragma unroll 8
        for (int k = 0; k < EE; ++k) acc += s_temb[nl * EE + k] * W_ih[(EE + k) * 128 + j];
        const float* hrow = h + (size_t)n * RR;
        #pragma unroll 8
        for (int r = 0; r < RR; ++r) acc += hrow[r] * W_hh[r * 128 + j];
        s_gates[nl * 128 + j] = acc;
    }
    __syncthreads();

    // Phase 3: LSTM pointwise  (64 nodes x 32 r)
    for (int w = tid; w < NB * RR; w += 256) {
        const int nl = w >> 5;
        const int r  = w & 31;
        const int n  = node0 + nl;

        const float gi = s_gates[nl * 128 + r];
        const float gf = s_gates[nl * 128 + 32 + r];
        const float gg = s_gates[nl * 128 + 64 + r];
        const float go = s_gates[nl * 128 + 96 + r];

        const float c_old = c[(size_t)n * RR + r];
        const float c_new = sigmoidf_(gf) * c_old + sigmoidf_(gi) * tanhf(gg);
        const float h_new = sigmoidf_(go) * tanhf(c_new);

        c[(size_t)n * RR + r] = c_new;
        h[(size_t)n * RR + r] = h_new;
        s_hnew[nl * RR + r] = h_new;
    }
    __syncthreads();

    // Phase 4: out = h_new @ W_out + b_out  (64 nodes x 5)
    for (int w = tid; w < NB * OO; w += 256) {
        const int nl = w / OO;
        const int o  = w % OO;
        const int n  = node0 + nl;
        float acc = b_out[o];
        #pragma unroll
        for (int r = 0; r < RR; ++r) acc += s_hnew[nl * RR + r] * W_out[r * OO + o];
        out_t[(size_t)n * OO + o] = acc;
    }
}

// ---------------------------------------------------------------------------
extern "C" void kernel_launch(void* const* d_in, const int* in_sizes, int n_in,
                              void* d_out, int out_size, void* d_ws, size_t ws_size,
                              hipStream_t stream) {
    const float* input_data = (const float*)d_in[0];
    const float* grids      = (const float*)d_in[1];
    const float* h0         = (const float*)d_in[2];
    const float* c0         = (const float*)d_in[3];
    const float* W_in       = (const float*)d_in[4];
    const float* b_in       = (const float*)d_in[5];
    const float* W_soc      = (const float*)d_in[6];
    const float* b_soc      = (const float*)d_in[7];
    const float* W_ih       = (const float*)d_in[8];
    const float* W_hh       = (const float*)d_in[9];
    const float* b_lstm     = (const float*)d_in[10];
    const float* W_out      = (const float*)d_in[11];
    const float* b_out      = (const float*)d_in[12];
    float* out = (float*)d_out;

    float* ws        = (float*)d_ws;
    float* s_part    = ws;                                        // SPLIT*N*512 floats (8 MB)
    float* s_social  = s_part + (size_t)SPLIT * NN * G2 * RR;     // N*512 floats (2 MB)
    float* s_h       = s_social + (size_t)NN * G2 * RR;           // N*32
    float* s_c       = s_h + (size_t)NN * RR;                     // N*32

    hipMemcpyAsync(s_h, h0, (size_t)NN * RR * sizeof(float), hipMemcpyDeviceToDevice, stream);
    hipMemcpyAsync(s_c, c0, (size_t)NN * RR * sizeof(float), hipMemcpyDeviceToDevice, stream);

    const size_t lds1 = (size_t)MCHUNK * RR * sizeof(float);      // 32 KB
    const size_t lds2 = (size_t)(512 * EE + 2 * NB * EE + NB * 128 + NB * RR) * sizeof(float); // 200 KB

    const dim3 grid1(NN / 16, SPLIT);   // 64 x 4 = 256 WGs, 16 waves each
    const int red_threads = 256;
    const int red_blocks  = (NN * G2 * RR / 4) / red_threads;     // 512 blocks of float4 work

    for (int t = 0; t < TT; ++t) {
        social_wmma_kernel<<<grid1, 512, lds1, stream>>>(
            grids + (size_t)t * NN * NN * G2, s_h, s_part);
        social_reduce_kernel<<<red_blocks, red_threads, 0, stream>>>(s_part, s_social);
        lstm_step_kernel<<<NN / NB, 256, lds2, stream>>>(
            input_data + (size_t)t * NN * 2, s_social, s_h, s_c,
            W_in, b_in, W_soc, b_soc, W_ih, W_hh, b_lstm, W_out, b_out,
            out + (size_t)t * NN * OO);
    }

    // final h, c appended after outputs [T,N,O]
    hipMemcpyAsync(out + (size_t)TT * NN * OO, s_h,
                   (size_t)NN * RR * sizeof(float), hipMemcpyDeviceToDevice, stream);
    hipMemcpyAsync(out + (size_t)TT * NN * OO + (size_t)NN * RR, s_c,
                   (size_t)NN * RR * sizeof(float), hipMemcpyDeviceToDevice, stream);
}